// Mahalanobis_mask_34419867910281
// MI455X (gfx1250) — compile-verified
//
#include <hip/hip_runtime.h>
#include <math.h>

// ---------------------------------------------------------------------------
// Problem: B=16, C=128, L=512, F=257
//   XF   = |rfft(X)|                     (2048 x 257)
//   P    = XF @ A^T                      (2048 x 257)   [= A applied to each spectrum]
//   dist[b,x,c] = || P[b,x] - P[b,c] ||^2
//   p    = 0.99 * (1/(dist+eps)) / rowmax   (diag = 0.99)
//   mask = hard gumbel sample: 1 iff log(p/(1-p))+g0 >= log((1-p)/p)+g1
// Padded dims: K 257->260 (mult of 4 for wmma f32 16x16x4), N 257->272 (mult of 16)
// ---------------------------------------------------------------------------

#define Lq   512
#define Fq   257
#define KP   260      // padded K (freq) dimension
#define NP   272      // padded N (output feature) dimension
#define ROWS 2048     // B*C
#define EPSq 1e-10f

typedef __attribute__((ext_vector_type(2))) float v2f;
typedef __attribute__((ext_vector_type(8))) float v8f;

// ---------------------------------------------------------------------------
// K1: XFp[row, f] = | DFT_f(X[row, :]) |  for f<257, zeros for 257<=f<260.
// One block per row; twiddle table in LDS, exact modular indexing.
// ---------------------------------------------------------------------------
__global__ __launch_bounds__(256)
void dft_mag_kernel(const float* __restrict__ X, float* __restrict__ XFp) {
    __shared__ float xs[Lq];
    __shared__ float ct[Lq];
    __shared__ float st[Lq];
    const int row = blockIdx.x;
    const int t = threadIdx.x;

    xs[t]       = X[row * Lq + t];
    xs[t + 256] = X[row * Lq + t + 256];
    for (int i = t; i < Lq; i += 256) {
        float s, c;
        sincosf(6.28318530717958647692f * (float)i / (float)Lq, &s, &c);
        ct[i] = c;
        st[i] = s;
    }
    __syncthreads();

    for (int f = t; f < KP; f += 256) {
        if (f < Fq) {
            float re = 0.f, im = 0.f;
            int idx = 0;
            #pragma unroll 4
            for (int l = 0; l < Lq; ++l) {
                const float xv = xs[l];
                re = fmaf(xv, ct[idx], re);
                im = fmaf(xv, st[idx], im);
                idx = (idx + f) & (Lq - 1);
            }
            XFp[row * KP + f] = sqrtf(fmaf(re, re, im * im));
        } else {
            XFp[row * KP + f] = 0.f;   // K padding
        }
    }
}

// ---------------------------------------------------------------------------
// K2: Bp[k, d] = A[d, k] (transpose) with zero padding to 260 x 272.
// ---------------------------------------------------------------------------
__global__ __launch_bounds__(256)
void build_B_kernel(const float* __restrict__ A, float* __restrict__ Bp) {
    const int i = blockIdx.x * 256 + threadIdx.x;
    if (i >= KP * NP) return;
    const int k = i / NP;
    const int d = i % NP;
    Bp[i] = (k < Fq && d < Fq) ? A[d * Fq + k] : 0.f;
}

// ---------------------------------------------------------------------------
// K3: P = XFp (2048 x 260) @ Bp (260 x 272) with V_WMMA_F32_16X16X4_F32.
// One 16x16 output tile per wave32; 8 waves per block; 272 blocks = 2176 tiles.
// Fragment layouts per CDNA5 ISA 7.12.2 (32-bit A 16x4, B 4x16, C/D 16x16).
// ---------------------------------------------------------------------------
__global__ __launch_bounds__(256)
void gemm_wmma_kernel(const float* __restrict__ XFp,
                      const float* __restrict__ Bp,
                      float* __restrict__ P) {
    const int lane = threadIdx.x & 31;
    const int wv   = threadIdx.x >> 5;
    const int tile = blockIdx.x * 8 + wv;        // 0..2175
    const int tm   = tile / 17;                  // 0..127  (M tiles)
    const int tn   = tile % 17;                  // 0..16   (N tiles)

    const int half = lane >> 4;                  // 0 | 1
    const int l16  = lane & 15;
    const int arow = tm * 16 + l16;              // A-fragment row (M)
    const int koff = half * 2;                   // k sub-offset: lanes 0-15 -> k0,k0+1 ; 16-31 -> k0+2,k0+3
    const int ncol = tn * 16 + l16;              // B/C column (N)

    const float* __restrict__ arow_p = XFp + arow * KP;

    v8f acc = {};
    for (int k0 = 0; k0 < KP; k0 += 4) {
        // hint-prefetch upcoming A data (lowers to global_prefetch_b8)
        __builtin_prefetch(arow_p + k0 + 32, 0, 1);
        v2f a, b;
        a.x = arow_p[k0 + koff];
        a.y = arow_p[k0 + koff + 1];
        b.x = Bp[(k0 + koff) * NP + ncol];
        b.y = Bp[(k0 + koff + 1) * NP + ncol];
        acc = __builtin_amdgcn_wmma_f32_16x16x4_f32(
                  /*neg_a=*/false, a, /*neg_b=*/false, b,
                  /*c_mod=*/(short)0, acc,
                  /*reuse_a=*/false, /*reuse_b=*/false);
    }

    // C/D layout: VGPR r -> M = r + 8*half, N = l16
    const int mrow0 = tm * 16 + half * 8;
    #pragma unroll
    for (int r = 0; r < 8; ++r) {
        P[(mrow0 + r) * NP + ncol] = acc[r];
    }
}

// ---------------------------------------------------------------------------
// K4: fused pairwise distance + row max + gumbel hard sample.
// One block per (b,x) row: 128 threads, thread c handles column c.
// Padded P columns are exactly zero -> loop full NP.
// ---------------------------------------------------------------------------
__global__ __launch_bounds__(128)
void dist_mask_kernel(const float* __restrict__ P,
                      const float* __restrict__ u,
                      float* __restrict__ out) {
    __shared__ float px[NP];
    __shared__ float red[128];

    const int rowx = blockIdx.x;          // b*128 + x
    const int x    = rowx & 127;
    const int b    = rowx >> 7;
    const int c    = threadIdx.x;

    for (int i = c; i < NP; i += 128) px[i] = P[rowx * NP + i];
    __syncthreads();

    const float* __restrict__ pc = P + ((b * 128 + c) * NP);
    float acc = 0.f;
    #pragma unroll 4
    for (int d = 0; d < NP; ++d) {
        const float v = px[d] - pc[d];
        acc = fmaf(v, v, acc);
    }

    // exp_dist with zeroed diagonal
    const float e = (c == x) ? 0.f : 1.f / (acc + EPSq);

    // block max over 128 threads
    red[c] = e;
    __syncthreads();
    for (int s = 64; s > 0; s >>= 1) {
        if (c < s) red[c] = fmaxf(red[c], red[c + s]);
        __syncthreads();
    }
    const float m = red[0];

    const float p  = (c == x) ? 0.99f : 0.99f * (e / (m + EPSq));
    const float l0 = logf(p / (1.f - p));          // logits[0]; logits[1] = -l0

    const long n  = (long)rowx * 128 + c;          // flat index into (B*C*C, 2)
    const float u0 = u[2 * n + 0];
    const float u1 = u[2 * n + 1];
    const float g0 = -logf(-logf(u0 + 1e-20f) + 1e-20f);
    const float g1 = -logf(-logf(u1 + 1e-20f) + 1e-20f);

    // argmax over {l0+g0, -l0+g1}; ties -> index 0 -> sample=1
    out[n] = (l0 + g0 >= -l0 + g1) ? 1.0f : 0.0f;
}

// ---------------------------------------------------------------------------
extern "C" void kernel_launch(void* const* d_in, const int* in_sizes, int n_in,
                              void* d_out, int out_size, void* d_ws, size_t ws_size,
                              hipStream_t stream) {
    (void)in_sizes; (void)n_in; (void)out_size; (void)ws_size;
    const float* X = (const float*)d_in[0];   // (16,128,512)
    const float* A = (const float*)d_in[1];   // (257,257)
    const float* u = (const float*)d_in[2];   // (262144,2)
    float* out = (float*)d_out;               // (16,1,128,128)

    float* XFp = (float*)d_ws;                // 2048*260 = 532480 floats
    float* P   = XFp + (size_t)ROWS * KP;     // 2048*272 = 557056 floats
    float* Bp  = P + (size_t)ROWS * NP;       // 260*272  =  70720 floats
                                              // total ~4.64 MB of workspace

    dft_mag_kernel<<<ROWS, 256, 0, stream>>>(X, XFp);
    build_B_kernel<<<(KP * NP + 255) / 256, 256, 0, stream>>>(A, Bp);
    gemm_wmma_kernel<<<272, 256, 0, stream>>>(XFp, Bp, P);
    dist_mask_kernel<<<ROWS, 128, 0, stream>>>(P, u, out);
}